// GCN_65231963292076
// MI455X (gfx1250) — compile-verified
//
#include <hip/hip_runtime.h>

#define NN 100000
#define EE 1600000
#define DH 128
#define DOUT2 40
#define BN_EPS 1e-5f
#define PITCH 132          // LDS pitch (floats) to avoid 64-bank conflicts
#define ROWT 5             // row tiles (of 16) per block: 80 rows, 100000 = 1250*80

typedef __attribute__((ext_vector_type(2))) float v2f;
typedef __attribute__((ext_vector_type(8))) float v8f;

// Native CDNA5 fp32 scatter-add: no-return atomic, device scope (RMW at L2).
__device__ __forceinline__ void atomic_add_f32(float* p, float v) {
    asm volatile("global_atomic_add_f32 %0, %1, off scope:SCOPE_DEV"
                 :: "v"(p), "v"(v) : "memory");
}

// ---- degree / normalization ------------------------------------------------
__global__ void deg_kernel(const int* __restrict__ dst, float* __restrict__ deg) {
    int e = blockIdx.x * blockDim.x + threadIdx.x;
    if (e < EE) atomic_add_f32(&deg[dst[e]], 1.0f);
}

__global__ void dinv_kernel(float* __restrict__ dinv) {
    int i = blockIdx.x * blockDim.x + threadIdx.x;
    if (i < NN) dinv[i] = rsqrtf(dinv[i] + 1.0f);  // +1 = self-loop
}

// ---- GEMM: out[N x Dout] = A[N x 128] @ W[128 x Dout], f32 WMMA 16x16x4 ----
// COLT col tiles of 16 (zero-padded past Dout). 80 rows per block (5 row tiles).
template <int Dout, int COLT>
__global__ __launch_bounds__(256) void gemm_wmma_kernel(
        const float* __restrict__ A, const float* __restrict__ W,
        float* __restrict__ out) {
    __shared__ float sA[16 * ROWT * PITCH];   // [row][k], pitch 132
    __shared__ float sW[16 * COLT * PITCH];   // transposed: [col][k], pitch 132

    const int rowBase = blockIdx.x * (16 * ROWT);

    // stage A tile: 80 rows x 128 k, coalesced float4 reads
    for (int q = threadIdx.x; q < 16 * ROWT * 32; q += 256) {
        int r = q >> 5, cc = (q & 31) << 2;
        float4 v = *(const float4*)&A[(rowBase + r) * DH + cc];
        *(float4*)&sA[r * PITCH + cc] = v;
    }
    // stage W transposed: consecutive threads -> consecutive cols (coalesced)
    for (int q = threadIdx.x; q < 16 * COLT * 32; q += 256) {
        int c = q % (16 * COLT);
        int k0 = (q / (16 * COLT)) << 2;
        float4 v;
        v.x = (c < Dout) ? W[(k0 + 0) * Dout + c] : 0.0f;
        v.y = (c < Dout) ? W[(k0 + 1) * Dout + c] : 0.0f;
        v.z = (c < Dout) ? W[(k0 + 2) * Dout + c] : 0.0f;
        v.w = (c < Dout) ? W[(k0 + 3) * Dout + c] : 0.0f;
        *(float4*)&sW[c * PITCH + k0] = v;
    }
    __syncthreads();

    const int wave = threadIdx.x >> 5;
    const int lane = threadIdx.x & 31;
    const int half = lane >> 4;      // lanes 16..31: upper half of K / M
    const int l    = lane & 15;

    for (int ct = wave; ct < COLT; ct += 8) {
        const int colL = ct * 16 + l;          // local col (0..COLT*16)
        v8f acc[ROWT];
        #pragma unroll
        for (int rt = 0; rt < ROWT; ++rt) acc[rt] = (v8f){};

        #pragma unroll 4
        for (int k0 = 0; k0 < DH; k0 += 4) {
            const int kk = k0 + 2 * half;
            v2f b;                             // B 4x16: V0=K(0|2), V1=K(1|3)
            b.x = sW[colL * PITCH + kk];
            b.y = sW[colL * PITCH + kk + 1];
            #pragma unroll
            for (int rt = 0; rt < ROWT; ++rt) {
                v2f a;                         // A 16x4: V0=K(0|2), V1=K(1|3)
                a.x = sA[(rt * 16 + l) * PITCH + kk];
                a.y = sA[(rt * 16 + l) * PITCH + kk + 1];
                acc[rt] = __builtin_amdgcn_wmma_f32_16x16x4_f32(
                    false, a, false, b, (short)0, acc[rt], false, false);
            }
        }

        if (colL < Dout) {                     // compile-time true for Dout=128
            #pragma unroll
            for (int rt = 0; rt < ROWT; ++rt) {
                #pragma unroll
                for (int v = 0; v < 8; ++v)    // C/D: row = v + 8*half
                    out[(rowBase + rt * 16 + v + 8 * half) * Dout + colL] = acc[rt][v];
            }
        }
    }
}

// ---- edge aggregation: out[dst] += tmp[src] * dinv[src]*dinv[dst] ----------
__global__ void agg_kernel(const float* __restrict__ tmp, const int* __restrict__ src,
                           const int* __restrict__ dst, const float* __restrict__ dinv,
                           float* __restrict__ out, int D, int chunks, int total) {
    int tid = blockIdx.x * blockDim.x + threadIdx.x;
    if (tid >= total) return;
    int e = tid / chunks;
    int c = (tid % chunks) << 2;
    int s = src[e], d = dst[e];
    float nrm = dinv[s] * dinv[d];
    float4 v = *(const float4*)&tmp[(size_t)s * D + c];
    float* o = &out[(size_t)d * D + c];
    atomic_add_f32(o + 0, v.x * nrm);
    atomic_add_f32(o + 1, v.y * nrm);
    atomic_add_f32(o + 2, v.z * nrm);
    atomic_add_f32(o + 3, v.w * nrm);
}

// ---- fused self-loop + bias + BN + ReLU (D = 128) --------------------------
__global__ void bnrelu_kernel(const float* __restrict__ agg, const float* __restrict__ tmp,
                              const float* __restrict__ dinv, const float* __restrict__ b,
                              const float* __restrict__ g, const float* __restrict__ beta,
                              const float* __restrict__ mean, const float* __restrict__ var,
                              float* __restrict__ out) {
    int idx = blockIdx.x * blockDim.x + threadIdx.x;
    if (idx >= NN * 32) return;
    int i = idx >> 5;
    int cc = (idx & 31) << 2;
    float di = dinv[i];
    float sl = di * di;
    float4 a  = *(const float4*)&agg[(size_t)i * DH + cc];
    float4 t  = *(const float4*)&tmp[(size_t)i * DH + cc];
    float4 bb = *(const float4*)&b[cc];
    float4 gg = *(const float4*)&g[cc];
    float4 be = *(const float4*)&beta[cc];
    float4 mm = *(const float4*)&mean[cc];
    float4 vv = *(const float4*)&var[cc];
    float4 r;
    r.x = fmaxf(gg.x * (a.x + t.x * sl + bb.x - mm.x) * rsqrtf(vv.x + BN_EPS) + be.x, 0.0f);
    r.y = fmaxf(gg.y * (a.y + t.y * sl + bb.y - mm.y) * rsqrtf(vv.y + BN_EPS) + be.y, 0.0f);
    r.z = fmaxf(gg.z * (a.z + t.z * sl + bb.z - mm.z) * rsqrtf(vv.z + BN_EPS) + be.z, 0.0f);
    r.w = fmaxf(gg.w * (a.w + t.w * sl + bb.w - mm.w) * rsqrtf(vv.w + BN_EPS) + be.w, 0.0f);
    *(float4*)&out[(size_t)i * DH + cc] = r;
}

// ---- fused self-loop + bias + log_softmax (D = 40), one wave per row -------
__global__ void lsm_kernel(const float* __restrict__ agg, const float* __restrict__ tmp,
                           const float* __restrict__ dinv, const float* __restrict__ b2,
                           float* __restrict__ out) {
    int row  = (blockIdx.x * blockDim.x + threadIdx.x) >> 5;
    int lane = threadIdx.x & 31;
    if (row >= NN) return;
    float di = dinv[row];
    float sl = di * di;
    int c0 = lane, c1 = lane + 32;
    float v0 = agg[(size_t)row * DOUT2 + c0] + tmp[(size_t)row * DOUT2 + c0] * sl + b2[c0];
    float v1 = -1e30f;
    if (c1 < DOUT2)
        v1 = agg[(size_t)row * DOUT2 + c1] + tmp[(size_t)row * DOUT2 + c1] * sl + b2[c1];
    float m = fmaxf(v0, v1);
    #pragma unroll
    for (int off = 16; off; off >>= 1) m = fmaxf(m, __shfl_xor(m, off, 32));
    float s = expf(v0 - m) + ((c1 < DOUT2) ? expf(v1 - m) : 0.0f);
    #pragma unroll
    for (int off = 16; off; off >>= 1) s += __shfl_xor(s, off, 32);
    float ls = logf(s);
    out[(size_t)row * DOUT2 + c0] = v0 - m - ls;
    if (c1 < DOUT2) out[(size_t)row * DOUT2 + c1] = v1 - m - ls;
}

// ---------------------------------------------------------------------------
extern "C" void kernel_launch(void* const* d_in, const int* in_sizes, int n_in,
                              void* d_out, int out_size, void* d_ws, size_t ws_size,
                              hipStream_t stream) {
    const float* x   = (const float*)d_in[0];
    const int*   ei  = (const int*)d_in[1];
    const int*   src = ei;
    const int*   dst = ei + EE;
    const float* W0  = (const float*)d_in[2];
    const float* b0  = (const float*)d_in[3];
    const float* W1  = (const float*)d_in[4];
    const float* b1  = (const float*)d_in[5];
    const float* W2  = (const float*)d_in[6];
    const float* b2  = (const float*)d_in[7];
    const float* g0  = (const float*)d_in[8];
    const float* be0 = (const float*)d_in[9];
    const float* m0  = (const float*)d_in[10];
    const float* v0  = (const float*)d_in[11];
    const float* g1  = (const float*)d_in[12];
    const float* be1 = (const float*)d_in[13];
    const float* m1  = (const float*)d_in[14];
    const float* v1  = (const float*)d_in[15];

    // workspace: dinv[N] | bufA[N*128] | bufB[N*128] | bufC[N*128]
    float* dinv = (float*)d_ws;
    float* bufA = dinv + 100032;                  // 64-aligned past N
    float* bufB = bufA + (size_t)NN * DH;
    float* bufC = bufB + (size_t)NN * DH;

    const int gemmBlocks = NN / (16 * ROWT);      // 1250
    const int aggTot128  = EE * 32;               // E * (128/4)
    const int aggTot40   = EE * 10;               // E * (40/4)

    // normalization
    hipMemsetAsync(dinv, 0, (size_t)NN * sizeof(float), stream);
    deg_kernel<<<(EE + 255) / 256, 256, 0, stream>>>(dst, dinv);
    dinv_kernel<<<(NN + 255) / 256, 256, 0, stream>>>(dinv);

    // layer 0: x @ W0 -> agg -> BN+ReLU
    gemm_wmma_kernel<DH, 8><<<gemmBlocks, 256, 0, stream>>>(x, W0, bufA);
    hipMemsetAsync(bufB, 0, (size_t)NN * DH * sizeof(float), stream);
    agg_kernel<<<(aggTot128 + 255) / 256, 256, 0, stream>>>(bufA, src, dst, dinv, bufB, DH, 32, aggTot128);
    bnrelu_kernel<<<(NN * 32 + 255) / 256, 256, 0, stream>>>(bufB, bufA, dinv, b0, g0, be0, m0, v0, bufC);

    // layer 1
    gemm_wmma_kernel<DH, 8><<<gemmBlocks, 256, 0, stream>>>(bufC, W1, bufA);
    hipMemsetAsync(bufB, 0, (size_t)NN * DH * sizeof(float), stream);
    agg_kernel<<<(aggTot128 + 255) / 256, 256, 0, stream>>>(bufA, src, dst, dinv, bufB, DH, 32, aggTot128);
    bnrelu_kernel<<<(NN * 32 + 255) / 256, 256, 0, stream>>>(bufB, bufA, dinv, b1, g1, be1, m1, v1, bufC);

    // layer 2 + log_softmax
    gemm_wmma_kernel<DOUT2, 3><<<gemmBlocks, 256, 0, stream>>>(bufC, W2, bufA);
    hipMemsetAsync(bufB, 0, (size_t)NN * DOUT2 * sizeof(float), stream);
    agg_kernel<<<(aggTot40 + 255) / 256, 256, 0, stream>>>(bufA, src, dst, dinv, bufB, DOUT2, 10, aggTot40);
    lsm_kernel<<<(NN + 7) / 8, 256, 0, stream>>>(bufB, bufA, dinv, b2, (float*)d_out);
}